// SparseMaskedMmOp_73710228734310
// MI455X (gfx1250) — compile-verified
//
#include <hip/hip_runtime.h>
#include <hip/hip_bf16.h>

typedef float v2f __attribute__((ext_vector_type(2)));
typedef float v8f __attribute__((ext_vector_type(8)));

#define NNZ_TOTAL 1000000
#define MDIM 8192
#define NDIM 8192
#define KDIM 128

// ---------------------------------------------------------------------------
// Kernel 1: transpose mat2 (K x N, row-major) -> mat2T (N x K, row-major).
// Classic LDS-tiled transpose with padding to avoid bank conflicts.
// 4 MB each way; runs once, makes every column gather a contiguous 512B read.
// ---------------------------------------------------------------------------
__global__ void transpose_kn_to_nk(const float* __restrict__ in, float* __restrict__ out) {
    __shared__ float tile[32][33];
    const int tx = threadIdx.x;          // 0..31
    const int ty = threadIdx.y;          // 0..7
    const int nBase = blockIdx.x * 32;   // N tile base
    const int kBase = blockIdx.y * 32;   // K tile base

    // Load 32x32 tile: rows = K, cols = N (coalesced along N)
    #pragma unroll
    for (int j = 0; j < 32; j += 8) {
        tile[ty + j][tx] = in[(long long)(kBase + ty + j) * NDIM + (nBase + tx)];
    }
    __syncthreads();
    // Store transposed: rows = N, cols = K (coalesced along K)
    #pragma unroll
    for (int j = 0; j < 32; j += 8) {
        out[(long long)(nBase + ty + j) * KDIM + (kBase + tx)] = tile[tx][ty + j];
    }
}

// ---------------------------------------------------------------------------
// Kernel 2: SDDMM via V_WMMA_F32_16X16X4_F32.
// Each wave handles 16 nnz: A = gathered mat1 rows (16xK), B = gathered
// mat2 columns (Kx16). Accumulate 16x16 = A*B over K=128 in 32 k-steps of 4,
// then take the diagonal, add mask_vals, store.
//
// fp32 WMMA operand layout (wave32):
//   A (16x4):  lane = m + 16*h (m=0..15, h=0..1), VGPR v holds A(m, 2h+v)
//   B (4x16):  lane = n + 16*h,                  VGPR v holds B(2h+v, n)
// -> per k-step both operands are a contiguous float2 at offset kk+2h of the
//    gathered row — one global_load_b64 per operand per step.
//   C (16x16): lanes 0..15: (M=v, N=lane); lanes 16..31: (M=8+v, N=lane-16)
// -> diagonal lives in lanes 0..7 (c[lane]) and lanes 24..31 (c[lane-24]).
// ---------------------------------------------------------------------------
template <bool B_IS_TRANSPOSED>
__global__ void sddmm_wmma(const float* __restrict__ mask_vals,
                           const long long* __restrict__ rows,
                           const long long* __restrict__ cols,
                           const float* __restrict__ mat1,
                           const float* __restrict__ matB,  // mat2T (NxK) or mat2 (KxN)
                           float* __restrict__ out) {
    const int lane = threadIdx.x & 31;
    const int wave = threadIdx.x >> 5;
    const int base = (blockIdx.x * 8 + wave) * 16;   // 16 nnz per wave

    const int m = lane & 15;   // which of the 16 nnz this lane's operand row is
    const int h = lane >> 4;   // K-half select within a k-step

    int idx = base + m;
    if (idx >= NNZ_TOTAL) idx = NNZ_TOTAL - 1;       // clamp tail (stores predicated)
    const int r = (int)rows[idx];
    const int c = (int)cols[idx];

    const float* __restrict__ arow = mat1 + (long long)r * KDIM + 2 * h;

    v8f acc = {};

    if (B_IS_TRANSPOSED) {
        const float* __restrict__ brow = matB + (long long)c * KDIM + 2 * h;
        #pragma unroll
        for (int kk = 0; kk < KDIM; kk += 4) {
            v2f a = *(const v2f*)(arow + kk);
            v2f b = *(const v2f*)(brow + kk);
            acc = __builtin_amdgcn_wmma_f32_16x16x4_f32(
                false, a, false, b, (short)0, acc, false, false);
        }
    } else {
        // Fallback: strided column reads of mat2 (K x N) if no workspace.
        #pragma unroll
        for (int kk = 0; kk < KDIM; kk += 4) {
            v2f a = *(const v2f*)(arow + kk);
            v2f b;
            b.x = matB[(long long)(kk + 2 * h)     * NDIM + c];
            b.y = matB[(long long)(kk + 2 * h + 1) * NDIM + c];
            acc = __builtin_amdgcn_wmma_f32_16x16x4_f32(
                false, a, false, b, (short)0, acc, false, false);
        }
    }

    // Extract diagonal element held by this lane (if any): sel = lane & 7.
    const int sel = lane & 7;
    float d = acc[0];
    d = (sel == 1) ? acc[1] : d;
    d = (sel == 2) ? acc[2] : d;
    d = (sel == 3) ? acc[3] : d;
    d = (sel == 4) ? acc[4] : d;
    d = (sel == 5) ? acc[5] : d;
    d = (sel == 6) ? acc[6] : d;
    d = (sel == 7) ? acc[7] : d;

    const bool holdsDiag = (lane < 8) || (lane >= 24);
    const int i = (lane < 8) ? lane : (lane - 16);   // diag index 0..15
    const int o = base + i;
    if (holdsDiag && o < NNZ_TOTAL) {
        out[o] = mask_vals[o] + d;
    }
}

extern "C" void kernel_launch(void* const* d_in, const int* in_sizes, int n_in,
                              void* d_out, int out_size, void* d_ws, size_t ws_size,
                              hipStream_t stream) {
    const float*     mask_vals = (const float*)d_in[0];
    const long long* rows      = (const long long*)d_in[1];   // int64 per reference
    const long long* cols      = (const long long*)d_in[2];   // int64 per reference
    const float*     mat1      = (const float*)d_in[3];       // M x K
    const float*     mat2      = (const float*)d_in[4];       // K x N
    float*           out       = (float*)d_out;

    const int nnz = in_sizes[0];                 // 1,000,000
    const int nnzPerBlock = 8 * 16;              // 8 waves x 16 nnz
    const int grid = (nnz + nnzPerBlock - 1) / nnzPerBlock;

    const size_t tNeeded = (size_t)NDIM * KDIM * sizeof(float);  // 4 MB
    if (ws_size >= tNeeded) {
        float* mat2T = (float*)d_ws;
        transpose_kn_to_nk<<<dim3(NDIM / 32, KDIM / 32), dim3(32, 8), 0, stream>>>(mat2, mat2T);
        sddmm_wmma<true><<<grid, 256, 0, stream>>>(mask_vals, rows, cols, mat1, mat2T, out);
    } else {
        sddmm_wmma<false><<<grid, 256, 0, stream>>>(mask_vals, rows, cols, mat1, mat2, out);
    }
}